// AttentionCombine_85375359910653
// MI455X (gfx1250) — compile-verified
//
#include <hip/hip_runtime.h>
#include <hip/hip_bf16.h>
#include <math.h>

typedef __attribute__((ext_vector_type(2))) float v2f;
typedef __attribute__((ext_vector_type(8))) float v8f;

#define N_CT   2048     // B*T contours
#define K1     2112     // 66*32
#define D_EMBD 512
#define D2     1024
#define T_SEQ  64
#define HW_DIM 160
#define CHN    64
#define P_PTS  32       // NUM_POINTS / STRIDE

// ---------------------------------------------------------------------------
// Kernel 1: bilinear gather -> cfA[n][c*32 + p], c in [0,66)
// block = (n,p) pair, 64 threads = channels; thread 0 also writes normed xy.
// ---------------------------------------------------------------------------
__global__ void gather_kernel(const float* __restrict__ cnn,
                              const float* __restrict__ contours,
                              const int*   __restrict__ img_idx,
                              const int*   __restrict__ hptr,
                              const int*   __restrict__ wptr,
                              float* __restrict__ cfA) {
  const int np = blockIdx.x;
  const int n  = np >> 5;
  const int p  = np & 31;
  const int c  = threadIdx.x;            // 0..63

  const float wim = (float)wptr[0];      // 640
  const float him = (float)hptr[0];      // 640
  const float x = contours[(size_t)n * 256 + p * 8 + 0];  // stride-4 point
  const float y = contours[(size_t)n * 256 + p * 8 + 1];

  const float px = x * ((float)HW_DIM / wim) - 0.5f;
  const float py = y * ((float)HW_DIM / him) - 0.5f;
  const float x0 = floorf(px), y0 = floorf(py);
  const float wx1 = px - x0, wx0 = 1.0f - wx1;   // (x0+1-px)
  const float wy1 = py - y0, wy0 = 1.0f - wy1;   // (y0+1-py)
  const int ix0 = (int)x0, iy0 = (int)y0;

  const int img = img_idx[n];
  const float* base = cnn + ((size_t)img * CHN + c) * (HW_DIM * HW_DIM);

  float acc = 0.0f;
  #pragma unroll
  for (int dy = 0; dy < 2; ++dy) {
    #pragma unroll
    for (int dx = 0; dx < 2; ++dx) {
      const int xi = ix0 + dx, yi = iy0 + dy;
      const bool valid = (xi >= 0) && (xi < HW_DIM) && (yi >= 0) && (yi < HW_DIM);
      int xc = xi < 0 ? 0 : (xi > HW_DIM - 1 ? HW_DIM - 1 : xi);
      int yc = yi < 0 ? 0 : (yi > HW_DIM - 1 ? HW_DIM - 1 : yi);
      const float wgt = (dx ? wx1 : wx0) * (dy ? wy1 : wy0);
      const float v = base[yc * HW_DIM + xc];
      acc += v * (valid ? wgt : 0.0f);
    }
  }
  cfA[(size_t)n * K1 + c * P_PTS + p] = acc;
  if (c == 0) {
    cfA[(size_t)n * K1 + 64 * P_PTS + p] = x / wim;
    cfA[(size_t)n * K1 + 65 * P_PTS + p] = y / him;
  }
}

// ---------------------------------------------------------------------------
// Register-blocked WMMA fp32 16x16x4 inner loop: one wave computes a 16x64
// output strip (4 accumulators). A float2 is reused by 4 WMMAs ->
// 1.25 loads per WMMA instead of 2.
// Bt is B stored transposed: B[k][n] = Bt[n*ldb + k].
// Lane layout (ISA 7.12.2, f32): lane = idx + 16*(k>=2); vgpr = k&1 ->
// each lane loads consecutive-k float2 pairs.
// ---------------------------------------------------------------------------
__device__ __forceinline__ void wmma_f32_dot4(const float* __restrict__ Arow,
                                              const float* __restrict__ B0,
                                              const float* __restrict__ B1,
                                              const float* __restrict__ B2,
                                              const float* __restrict__ B3,
                                              int K, v8f acc[4]) {
  #pragma unroll 4
  for (int k = 0; k < K; k += 4) {
    v2f a  = *(const v2f*)(Arow + k);
    v2f b0 = *(const v2f*)(B0 + k);
    v2f b1 = *(const v2f*)(B1 + k);
    v2f b2 = *(const v2f*)(B2 + k);
    v2f b3 = *(const v2f*)(B3 + k);
    acc[0] = __builtin_amdgcn_wmma_f32_16x16x4_f32(false, a, false, b0, (short)0, acc[0], false, false);
    acc[1] = __builtin_amdgcn_wmma_f32_16x16x4_f32(false, a, false, b1, (short)0, acc[1], false, false);
    acc[2] = __builtin_amdgcn_wmma_f32_16x16x4_f32(false, a, false, b2, (short)0, acc[2], false, false);
    acc[3] = __builtin_amdgcn_wmma_f32_16x16x4_f32(false, a, false, b3, (short)0, acc[3], false, false);
  }
}

// ---------------------------------------------------------------------------
// Kernel 2: cf1[2048][512] = cfA[2048][2112] x conv_w^T + conv_b + pos_embed
// conv_w is (512,66,32) row-major == Bt with ldb = 2112.
// One wave per 16x64 strip; strips = 128 rows x 8 col-groups = 1024 waves.
// ---------------------------------------------------------------------------
__global__ void gemm1_kernel(const float* __restrict__ cfA,
                             const float* __restrict__ convw,
                             const float* __restrict__ convb,
                             const float* __restrict__ pos,
                             const int*   __restrict__ ct_ind,
                             float* __restrict__ cf1) {
  const int wave = blockIdx.x * (blockDim.x >> 5) + (threadIdx.x >> 5);
  const int tr  = wave >> 3;            // 0..127
  const int tcg = wave & 7;             // 0..7   (col group of 64)
  const int lane = threadIdx.x & 31;
  const int half = lane >> 4;
  const int mr   = lane & 15;
  const int row0 = tr * 16, col0 = tcg * 64;

  const float* Arow = cfA   + (size_t)(row0 + mr) * K1 + 2 * half;
  const float* B0   = convw + (size_t)(col0 + mr) * K1 + 2 * half;
  v8f acc[4] = {{0.f},{0.f},{0.f},{0.f}};
  wmma_f32_dot4(Arow, B0, B0 + 16 * K1, B0 + 32 * K1, B0 + 48 * K1, K1, acc);

  #pragma unroll
  for (int g = 0; g < 4; ++g) {
    const int col = col0 + g * 16 + mr;
    const float bias = convb[col];
    #pragma unroll
    for (int r = 0; r < 8; ++r) {
      const int row = row0 + r + 8 * half;
      const int ci = ct_ind[row];
      const int cx = ((ci % HW_DIM) * 16) / HW_DIM;
      const int cy = ((ci / HW_DIM) * 16) / HW_DIM;
      cf1[(size_t)row * D_EMBD + col] = acc[g][r] + bias + pos[col * 256 + cy * 16 + cx];
    }
  }
}

// ---------------------------------------------------------------------------
// Kernel 3: qk[2048][1024] = cf1[2048][512] x attn_w^T + attn_b
// attn_w (1024,512) row-major == Bt with ldb = 512.
// Strips = 128 rows x 16 col-groups = 2048 waves.
// ---------------------------------------------------------------------------
__global__ void gemm2_kernel(const float* __restrict__ cf1,
                             const float* __restrict__ attnw,
                             const float* __restrict__ attnb,
                             float* __restrict__ qk) {
  const int wave = blockIdx.x * (blockDim.x >> 5) + (threadIdx.x >> 5);
  const int tr  = wave >> 4;            // 0..127
  const int tcg = wave & 15;            // 0..15
  const int lane = threadIdx.x & 31;
  const int half = lane >> 4;
  const int mr   = lane & 15;
  const int row0 = tr * 16, col0 = tcg * 64;

  const float* Arow = cf1   + (size_t)(row0 + mr) * D_EMBD + 2 * half;
  const float* B0   = attnw + (size_t)(col0 + mr) * D_EMBD + 2 * half;
  v8f acc[4] = {{0.f},{0.f},{0.f},{0.f}};
  wmma_f32_dot4(Arow, B0, B0 + 16 * D_EMBD, B0 + 32 * D_EMBD, B0 + 48 * D_EMBD,
                D_EMBD, acc);

  #pragma unroll
  for (int g = 0; g < 4; ++g) {
    const int col = col0 + g * 16 + mr;
    const float bias = attnb[col];
    #pragma unroll
    for (int r = 0; r < 8; ++r) {
      const int row = row0 + r + 8 * half;
      qk[(size_t)row * D2 + col] = acc[g][r] + bias;
    }
  }
}

// ---------------------------------------------------------------------------
// Kernel 4: serial inclusive scan of ct_01 -> order[i] = clip(cumsum-1, 0, N-1)
// (ct_01 is all-true in setup; generic anyway. jax bool = 1 byte.)
// ---------------------------------------------------------------------------
__global__ void scan_kernel(const unsigned char* __restrict__ ct01,
                            int* __restrict__ order) {
  if (threadIdx.x == 0 && blockIdx.x == 0) {
    int c = 0;
    for (int i = 0; i < N_CT; ++i) {
      c += (ct01[i] != 0) ? 1 : 0;
      int o = c - 1;
      o = o < 0 ? 0 : (o > N_CT - 1 ? N_CT - 1 : o);
      order[i] = o;
    }
  }
}

// ---------------------------------------------------------------------------
// Kernel 5: pack Qs/Ks with mask+order; fold p_w[head]/sqrt(hd) into Q so the
// head-combine becomes one 512-deep GEMM.
// ---------------------------------------------------------------------------
__global__ void pack_kernel(const float* __restrict__ qk,
                            const unsigned char* __restrict__ ct01,
                            const int* __restrict__ order,
                            const float* __restrict__ pw,
                            float* __restrict__ Qs,
                            float* __restrict__ Ks) {
  const int idx = blockIdx.x * blockDim.x + threadIdx.x;  // over 2048*512
  const int i = idx >> 9;
  const int j = idx & 511;
  const bool m = ct01[i] != 0;
  const int o = order[i];
  const float qv = m ? qk[(size_t)o * D2 + j] : 0.0f;
  const float kv = m ? qk[(size_t)o * D2 + D_EMBD + j] : 0.0f;
  Qs[idx] = qv * pw[j >> 6] * 0.125f;   // p_w[head] * rsqrt(64)
  Ks[idx] = kv;
}

// ---------------------------------------------------------------------------
// Kernel 6: out[b][t][s] = clean(sigmoid( Qs[b] x Ks[b]^T )), 32 batches of
// (64x512)x(512x64). One wave per 16x64 strip -> 32*4 = 128 waves.
// ---------------------------------------------------------------------------
__global__ void attn_kernel(const float* __restrict__ Qs,
                            const float* __restrict__ Ks,
                            float* __restrict__ out) {
  const int wave = blockIdx.x * (blockDim.x >> 5) + (threadIdx.x >> 5);
  const int b  = wave >> 2;             // 0..31
  const int tr = wave & 3;              // 0..3
  const int lane = threadIdx.x & 31;
  const int half = lane >> 4;
  const int mr   = lane & 15;
  const int row0 = tr * 16;

  const float* Arow = Qs + ((size_t)b * T_SEQ + row0 + mr) * D_EMBD + 2 * half;
  const float* B0   = Ks + ((size_t)b * T_SEQ + mr) * D_EMBD + 2 * half;
  v8f acc[4] = {{0.f},{0.f},{0.f},{0.f}};
  wmma_f32_dot4(Arow, B0, B0 + 16 * D_EMBD, B0 + 32 * D_EMBD, B0 + 48 * D_EMBD,
                D_EMBD, acc);

  #pragma unroll
  for (int g = 0; g < 4; ++g) {
    const int col = g * 16 + mr;
    #pragma unroll
    for (int r = 0; r < 8; ++r) {
      const int row = row0 + r + 8 * half;
      const float v = acc[g][r];
      float s = 1.0f / (1.0f + expf(-v));
      if (s != s) s = 0.0f;             // NaN -> 0
      out[(size_t)b * (T_SEQ * T_SEQ) + row * T_SEQ + col] = s;
    }
  }
}

// ---------------------------------------------------------------------------
extern "C" void kernel_launch(void* const* d_in, const int* in_sizes, int n_in,
                              void* d_out, int out_size, void* d_ws, size_t ws_size,
                              hipStream_t stream) {
  const float* cnn      = (const float*)d_in[0];           // (32,64,160,160)
  const float* contours = (const float*)d_in[1];           // (2048,128,2)
  const unsigned char* ct01 = (const unsigned char*)d_in[2]; // (32,64) bool
  const int*   img_idx  = (const int*)d_in[3];             // (2048,)
  const int*   ct_ind   = (const int*)d_in[4];             // (2048,)
  const int*   hptr     = (const int*)d_in[5];             // scalar 640
  const int*   wptr     = (const int*)d_in[6];             // scalar 640
  const float* convw    = (const float*)d_in[7];           // (512,66,32)
  const float* convb    = (const float*)d_in[8];           // (512,)
  const float* attnw    = (const float*)d_in[9];           // (1024,512)
  const float* attnb    = (const float*)d_in[10];          // (1024,)
  const float* pw       = (const float*)d_in[11];          // (1,8,1)
  const float* pos      = (const float*)d_in[12];          // (512,16,16)
  float* out = (float*)d_out;                              // (32,64,64)

  // Workspace carve-up (floats): total ~38.3 MB
  float* ws  = (float*)d_ws;
  float* cfA = ws;                                 // 2048*2112 = 4,325,376
  float* cf1 = cfA + (size_t)N_CT * K1;            // 2048*512  = 1,048,576
  float* qk  = cf1 + (size_t)N_CT * D_EMBD;        // 2048*1024 = 2,097,152
  float* Qs  = qk  + (size_t)N_CT * D2;            // 1,048,576
  float* Ks  = Qs  + (size_t)N_CT * D_EMBD;        // 1,048,576
  int*   order = (int*)(Ks + (size_t)N_CT * D_EMBD);

  // 1) gather: 2048*32 (n,p) blocks x 64 channel threads
  gather_kernel<<<N_CT * P_PTS, CHN, 0, stream>>>(cnn, contours, img_idx, hptr, wptr, cfA);

  // 2) GEMM1: 128x8 strips, 4 waves/block
  gemm1_kernel<<<(128 * 8) / 4, 128, 0, stream>>>(cfA, convw, convb, pos, ct_ind, cf1);

  // 3) GEMM2: 128x16 strips
  gemm2_kernel<<<(128 * 16) / 4, 128, 0, stream>>>(cf1, attnw, attnb, qk);

  // 4) order scan
  scan_kernel<<<1, 32, 0, stream>>>(ct01, order);

  // 5) pack Q'/K
  pack_kernel<<<(N_CT * D_EMBD) / 256, 256, 0, stream>>>(qk, ct01, order, pw, Qs, Ks);

  // 6) batched attention GEMM + sigmoid: 32x4 strips
  attn_kernel<<<(32 * 4) / 4, 128, 0, stream>>>(Qs, Ks, out);
}